// EigenvalueLayerNorm_3547642987273
// MI455X (gfx1250) — compile-verified
//
#include <hip/hip_runtime.h>
#include <math.h>

typedef __attribute__((ext_vector_type(2))) float v2f;
typedef __attribute__((ext_vector_type(4))) float v4f;
typedef __attribute__((ext_vector_type(8))) float v8f;

#define B_  8
#define F_  64
#define N_  256
#define EPS 1e-9f

// ---------------------------------------------------------------------------
// Pass 1: per-(b,f) masked trace(A) and masked trace(A^2) using WMMA f32
// One workgroup (256 threads = 8 waves) per (b,f) matrix.
// trace_sq = sum_{i,k} m[i]^2 * A[i,k] * A[k,i]
//          = sum over diagonal 16x16 blocks I of diag( sum_K T(I,K) @ T(K,I) )
//            weighted by m^2  -> computed with V_WMMA_F32_16X16X4_F32 chains.
// Reads of x here use default (RT) hints on purpose: they prime L2 so the
// apply pass reads hit in the 192 MB L2 instead of HBM.
// ---------------------------------------------------------------------------
__global__ __launch_bounds__(256) void eln_stats_kernel(
    const float* __restrict__ x,     // [B,F,N,N]
    const float* __restrict__ mask,  // [B,N]
    float* __restrict__ trace,       // [B*F]
    float* __restrict__ trace_sq)    // [B*F]
{
  const int bf = blockIdx.x;                   // 0..511
  const int b  = bf / F_;
  const float* __restrict__ A = x + (size_t)bf * N_ * N_;
  const float* __restrict__ m = mask + b * N_;

  const int tid  = threadIdx.x;
  const int wave = tid >> 5;
  const int lane = tid & 31;
  const int lo16 = lane & 15;
  const int hi   = lane >> 4;                  // 0 or 1 (half-wave)

  __shared__ float s_tr[8];
  __shared__ float s_ts[8];

  // ---- masked trace(A): thread t handles diagonal element t ----
  {
    float mt = m[tid];
    float v  = A[(size_t)tid * N_ + tid] * mt * mt;
    #pragma unroll
    for (int off = 16; off > 0; off >>= 1)
      v += __shfl_down(v, off, 32);
    if (lane == 0) s_tr[wave] = v;
  }

  // ---- masked trace(A^2) via WMMA on the 16 diagonal 16x16 blocks ----
  float ts_lane = 0.0f;
  for (int d = wave; d < 16; d += 8) {         // each wave owns 2 diag blocks
    const int I0 = d * 16;
    v8f acc = {};
    for (int K0 = 0; K0 < N_; K0 += 16) {
      #pragma unroll
      for (int c = 0; c < 16; c += 4) {
        // A-matrix fragment (16x4 f32): rows I0..I0+15, cols K0+c..K0+c+3
        //   lanes 0-15: K = c+0 (v0), c+1 (v1); lanes 16-31: K = c+2, c+3
        const int arow = I0 + lo16;
        const int acol = K0 + c + (hi << 1);
        v2f afrag = *(const v2f*)(A + (size_t)arow * N_ + acol);

        // B-matrix fragment (4x16 f32): rows K0+c..K0+c+3, cols I0..I0+15
        //   v0: rows c+0 (lanes 0-15) / c+2 (lanes 16-31); v1: rows c+1 / c+3
        const int bcol = I0 + lo16;
        const int brow = K0 + c + (hi << 1);
        v2f bfrag;
        bfrag.x = A[(size_t)brow * N_ + bcol];
        bfrag.y = A[(size_t)(brow + 1) * N_ + bcol];

        acc = __builtin_amdgcn_wmma_f32_16x16x4_f32(
            /*neg_a=*/false, afrag, /*neg_b=*/false, bfrag,
            /*c_mod=*/(short)0, acc, /*reuse_a=*/false, /*reuse_b=*/false);
      }
    }
    // Extract diagonal of the 16x16 product from the C/D layout:
    //   VGPR r: lanes 0-15 -> (M=r, N=lane); lanes 16-31 -> (M=r+8, N=lane-16)
    //   P[i,i]: i<8 at (vgpr=i, lane=i); i>=8 at (vgpr=i-8, lane=i+16)
    const int r = lane & 7;
    float dval = acc[0];
    dval = (r == 1) ? acc[1] : dval;
    dval = (r == 2) ? acc[2] : dval;
    dval = (r == 3) ? acc[3] : dval;
    dval = (r == 4) ? acc[4] : dval;
    dval = (r == 5) ? acc[5] : dval;
    dval = (r == 6) ? acc[6] : dval;
    dval = (r == 7) ? acc[7] : dval;
    const bool valid = (lane < 8) || (lane >= 24);
    const int  i     = (lane < 16) ? lane : (lane - 16);  // diag index in block
    const float mi   = valid ? m[I0 + i] : 0.0f;
    ts_lane += dval * mi * mi;
  }
  #pragma unroll
  for (int off = 16; off > 0; off >>= 1)
    ts_lane += __shfl_down(ts_lane, off, 32);
  if (lane == 0) s_ts[wave] = ts_lane;

  __syncthreads();
  if (tid == 0) {
    float tr = 0.0f, ts = 0.0f;
    #pragma unroll
    for (int w = 0; w < 8; ++w) { tr += s_tr[w]; ts += s_ts[w]; }  // fixed order
    trace[bf]    = tr;
    trace_sq[bf] = ts;
  }
}

// ---------------------------------------------------------------------------
// Pass 2: per-batch mean_b / var_b (8 blocks, deterministic LDS tree reduce)
// ---------------------------------------------------------------------------
__global__ __launch_bounds__(256) void eln_reduce_kernel(
    const float* __restrict__ mask,       // [B,N]
    const float* __restrict__ trace,      // [B*F]
    const float* __restrict__ trace_sq,   // [B*F]
    float* __restrict__ mean_b,           // [B]
    float* __restrict__ var_b)            // [B]
{
  const int b = blockIdx.x;
  const int t = threadIdx.x;
  __shared__ float red[256];
  __shared__ float n_sh;

  float mv = mask[b * N_ + t];
  red[t] = mv * mv;
  __syncthreads();
  #pragma unroll
  for (int s = 128; s > 0; s >>= 1) {
    if (t < s) red[t] += red[t + s];
    __syncthreads();
  }
  if (t == 0) n_sh = red[0];
  __syncthreads();
  const float n  = n_sh;
  const float n2 = fmaxf(n - 1.0f, 1.0f);

  float mean = 0.0f, var = 0.0f;
  if (t < F_) {
    const float tr = trace[b * F_ + t];
    const float ts = trace_sq[b * F_ + t];
    mean = tr / n;
    var  = ts / n2 - tr * tr / (n * n2);
  }
  red[t] = mean;
  __syncthreads();
  #pragma unroll
  for (int s = 128; s > 0; s >>= 1) {
    if (t < s) red[t] += red[t + s];
    __syncthreads();
  }
  if (t == 0) mean_b[b] = red[0] / (float)F_;
  __syncthreads();
  red[t] = var;
  __syncthreads();
  #pragma unroll
  for (int s = 128; s > 0; s >>= 1) {
    if (t < s) red[t] += red[t + s];
    __syncthreads();
  }
  if (t == 0) var_b[b] = red[0] / (float)F_;
}

// ---------------------------------------------------------------------------
// Pass 3: elementwise apply, float4-vectorized (bandwidth-bound).
// x loads are non-temporal (each element read exactly once; lines were primed
// into L2 by pass 1 and can be dropped after this read).  out stores are
// non-temporal so 128 MiB of output doesn't evict not-yet-read x from L2.
// out = ((x - mu*delta_ij) * m_i*m_j) * rsqrt(var+eps) * (w*e^we + wb)
//       + bias_f * delta_ij
// ---------------------------------------------------------------------------
__global__ __launch_bounds__(256) void eln_apply_kernel(
    const v4f*   __restrict__ x4,
    const float* __restrict__ mask,
    const float* __restrict__ weight,
    const float* __restrict__ weight_exp,
    const float* __restrict__ weight_bias,
    const float* __restrict__ bias,
    const float* __restrict__ mean_b,
    const float* __restrict__ var_b,
    v4f* __restrict__ out4)
{
  const int tid = blockIdx.x * blockDim.x + threadIdx.x;  // over N_/4 chunks
  // tid = ((b*F + f)*N + i) * (N/4) + j4
  const int j4 = tid & 63;          // N_/4 - 1
  int tmp      = tid >> 6;
  const int i  = tmp & (N_ - 1);
  tmp >>= 8;
  const int f  = tmp & (F_ - 1);
  const int b  = tmp >> 6;

  const v4f xv = __builtin_nontemporal_load(&x4[tid]);
  const float* __restrict__ mrow = mask + b * N_;
  const float  mi = mrow[i];
  const v4f mj = *(const v4f*)(mrow + (j4 << 2));

  const float mu  = mean_b[b];
  const float inv = rsqrtf(var_b[b] + EPS);
  const float sc  = (weight[f] * expf(weight_exp[f]) + weight_bias[f]) * inv;
  const float bs  = bias[f];

  const int j0 = j4 << 2;
  v4f o;
  {
    const bool dg = (i == j0 + 0);
    o.x = (xv.x - (dg ? mu : 0.0f)) * (mi * mj.x) * sc + (dg ? bs : 0.0f);
  }
  {
    const bool dg = (i == j0 + 1);
    o.y = (xv.y - (dg ? mu : 0.0f)) * (mi * mj.y) * sc + (dg ? bs : 0.0f);
  }
  {
    const bool dg = (i == j0 + 2);
    o.z = (xv.z - (dg ? mu : 0.0f)) * (mi * mj.z) * sc + (dg ? bs : 0.0f);
  }
  {
    const bool dg = (i == j0 + 3);
    o.w = (xv.w - (dg ? mu : 0.0f)) * (mi * mj.w) * sc + (dg ? bs : 0.0f);
  }
  __builtin_nontemporal_store(o, &out4[tid]);
}

// ---------------------------------------------------------------------------
extern "C" void kernel_launch(void* const* d_in, const int* in_sizes, int n_in,
                              void* d_out, int out_size, void* d_ws, size_t ws_size,
                              hipStream_t stream) {
  const float* x           = (const float*)d_in[0];  // [B,F,N,N]
  const float* mask        = (const float*)d_in[1];  // [B,N]
  const float* weight      = (const float*)d_in[2];  // [1,F,1,1]
  const float* weight_exp  = (const float*)d_in[3];  // [1,F,1,1]
  const float* weight_bias = (const float*)d_in[4];  // [1,F,1,1]
  const float* bias        = (const float*)d_in[5];  // [1,F,1]

  float* ws       = (float*)d_ws;
  float* trace    = ws;          // 512
  float* trace_sq = ws + 512;    // 512
  float* mean_b   = ws + 1024;   // 8
  float* var_b    = ws + 1032;   // 8

  eln_stats_kernel<<<B_ * F_, 256, 0, stream>>>(x, mask, trace, trace_sq);
  eln_reduce_kernel<<<B_, 256, 0, stream>>>(mask, trace, trace_sq, mean_b, var_b);

  const int total4 = (B_ * F_ * N_ * N_) / 4;        // 8,388,608
  eln_apply_kernel<<<total4 / 256, 256, 0, stream>>>(
      (const v4f*)x, mask, weight, weight_exp, weight_bias, bias,
      mean_b, var_b, (v4f*)d_out);
}